// KDABlock_4475355922517
// MI455X (gfx1250) — compile-verified
//
#include <hip/hip_runtime.h>

// ---------------------------------------------------------------------------
// KDA chunked gated delta-rule attention for MI455X (gfx1250, wave32, WMMA).
// Grid = (B*H) x 2 V-halves: the recurrent state's V-columns are independent,
// so splitting V doubles WGP occupancy at the cost of duplicating the k-side
// GEMMs (~17% extra FLOPs). Each workgroup scans its 32 chunks sequentially
// with its K x V/2 state resident in LDS; all GEMMs use V_WMMA_F32_16X16X4_F32
// (f32 matrix pipe -- exp-gated operands span ~1e+-14, so low-precision WMMA
// inputs are numerically unusable). Next-chunk inputs are prefetched into L2
// (global_prefetch_b8) to hide the chunk-boundary HBM latency in the scan.
// ---------------------------------------------------------------------------

typedef float v2f __attribute__((ext_vector_type(2)));
typedef float v8f __attribute__((ext_vector_type(8)));

#define LCH 64      // chunk length
#define DK  128     // key dim
#define DV  128     // full value dim
#define DVB 64      // value-dim block per workgroup
#define NVS 2       // V splits (DV / DVB)
#define NCH 32      // chunks per sequence (S / LCH)
#define BHN 32      // B * H
#define SEQ 2048
#define NTHREADS 256

__device__ __forceinline__ v8f wmma4(v2f a, v2f b, v8f c) {
  // D = A(16x4) * B(4x16) + C, full fp32
  return __builtin_amdgcn_wmma_f32_16x16x4_f32(
      /*neg_a=*/false, a, /*neg_b=*/false, b,
      /*c_mod=*/(short)0, c, /*reuse_a=*/false, /*reuse_b=*/false);
}

__global__ __launch_bounds__(NTHREADS) void kda_chunk_kernel(
    const float* __restrict__ q, const float* __restrict__ k,
    const float* __restrict__ v, const float* __restrict__ g,
    const float* __restrict__ beta, float* __restrict__ out,
    float* __restrict__ state_out) {
  // ---------------- LDS (~258 KB of the 320 KB WGP pool) ----------------
  __shared__ __attribute__((aligned(16))) float sS [DK * DVB];  // running state (32 KB)
  __shared__ __attribute__((aligned(16))) float sQ [LCH * DK];  // normalized q
  __shared__ __attribute__((aligned(16))) float sKD[LCH * DK];  // k * exp(-gcum)
  __shared__ __attribute__((aligned(16))) float sV [LCH * DVB]; // v * exp(gmean)
  __shared__ __attribute__((aligned(16))) float sCR[LCH * DK];  // g -> gcum; then U/corr (LCH*DVB)
  __shared__ __attribute__((aligned(16))) float sKG[LCH * DK];  // k * exp(+gcum)
  __shared__ __attribute__((aligned(16))) float sQD[LCH * DK];  // q * exp(-gcum) -> W
  __shared__ __attribute__((aligned(16))) float sA [LCH * LCH]; // I + beta * tril(kk,-1)
  __shared__ __attribute__((aligned(16))) float sT [LCH * LCH]; // A^-1 diag(beta)
  __shared__ __attribute__((aligned(16))) float sQK[LCH * LCH]; // causal qk * scale
  __shared__ float sGm[LCH];      // exp(mean_k gcum)
  __shared__ float sBeta[LCH];
  __shared__ float sDecay[DK];    // exp(g_total)
  __shared__ float sNrm[2 * LCH];

  const int tid  = threadIdx.x;
  const int lane = tid & 31;
  const int wave = tid >> 5;
  const int bh   = blockIdx.x >> 1;   // (b,h) pair
  const int vh   = blockIdx.x & 1;    // which V half
  const float scale = 0.08838834764831845f;  // K^-0.5

  const float* qb = q    + (size_t)bh * SEQ * DK;
  const float* kb = k    + (size_t)bh * SEQ * DK;
  const float* vb = v    + (size_t)bh * SEQ * DV + vh * DVB;
  const float* gb = g    + (size_t)bh * SEQ * DK;
  const float* bb = beta + (size_t)bh * SEQ;
  float*       ob = out  + (size_t)bh * SEQ * DV + vh * DVB;

  // zero the carried state
  for (int i = tid; i < DK * DVB; i += NTHREADS) sS[i] = 0.0f;
  __syncthreads();

  for (int c = 0; c < NCH; ++c) {
    // ---- stage chunk into LDS (float4 vector loads) ----
    const float* qp = qb + (size_t)c * LCH * DK;
    const float* kp = kb + (size_t)c * LCH * DK;
    const float* vp = vb + (size_t)c * LCH * DV;
    const float* gp = gb + (size_t)c * LCH * DK;
    for (int i = 4 * tid; i < LCH * DK; i += 4 * NTHREADS) {
      *(float4*)(sQ  + i) = *(const float4*)(qp + i);
      *(float4*)(sKD + i) = *(const float4*)(kp + i);
      *(float4*)(sCR + i) = *(const float4*)(gp + i);
    }
    for (int i = 4 * tid; i < LCH * DVB; i += 4 * NTHREADS) {
      int r = i >> 6, col = i & (DVB - 1);
      *(float4*)(sV + i) = *(const float4*)(vp + r * DV + col);
    }
    if (tid < LCH) sBeta[tid] = bb[c * LCH + tid];
    // ---- prefetch next chunk into L2 while this chunk computes ----
    if (c + 1 < NCH) {
      for (int i = 32 * tid; i < LCH * DK; i += 32 * NTHREADS) {
        __builtin_prefetch(qp + LCH * DK + i, 0, 1);
        __builtin_prefetch(kp + LCH * DK + i, 0, 1);
        __builtin_prefetch(gp + LCH * DK + i, 0, 1);
        __builtin_prefetch(vp + LCH * DV + i, 0, 1);
      }
    }
    __syncthreads();

    // ---- row norms (threads 0..127) and gcum columns (threads 128..255) ----
    if (tid < 2 * LCH) {
      int r = tid & (LCH - 1);
      const float* src = (tid < LCH) ? (sQ + r * DK) : (sKD + r * DK);
      float s = 0.0f;
      for (int kk = 0; kk < DK; ++kk) { float a = src[kk]; s += a * a; }
      sNrm[tid] = 1.0f / (sqrtf(s) + 1e-6f);
    } else {
      int col = tid - 2 * LCH;           // 0..127
      float s = 0.0f;
      for (int r = 0; r < LCH; ++r) { s += sCR[r * DK + col]; sCR[r * DK + col] = s; }
      sDecay[col] = expf(s);             // exp(g_total)
    }
    __syncthreads();

    // ---- gmean per row + apply normalization ----
    if (tid < LCH) {
      float s = 0.0f;
      for (int kk = 0; kk < DK; ++kk) s += sCR[tid * DK + kk];
      sGm[tid] = expf(s * (1.0f / DK));
    }
    for (int i = tid; i < LCH * DK; i += NTHREADS) {
      int r = i >> 7;
      sQ[i]  *= sNrm[r];
      sKD[i] *= sNrm[LCH + r];
    }
    __syncthreads();

    // ---- gating: kg = k e^{+gc}, kd = k e^{-gc}, qd = q e^{-gc}, vg = v e^{gmean} ----
    for (int i = tid; i < LCH * DK; i += NTHREADS) {
      float gc  = sCR[i];
      float eg  = expf(gc);
      float egn = expf(-gc);
      float kn  = sKD[i];
      sKG[i] = kn * eg;
      sKD[i] = kn * egn;
      sQD[i] = sQ[i] * egn;
    }
    for (int i = tid; i < LCH * DVB; i += NTHREADS)
      sV[i] *= sGm[i >> 6];
    __syncthreads();

    // ---- GEMM 1: A = I + beta*tril(kd.kg^T,-1); qk = causal(qd.kg^T)*scale ----
    // kg acts as B^T (contiguous float2 fragments); shared B-frag for both.
    for (int t = wave; t < 16; t += 8) {
      int tm = t >> 2, tn = t & 3;
      int r = lane & 15, h = lane >> 4;
      const float* ap = sKD + (tm * 16 + r) * DK + 2 * h;
      const float* aq = sQD + (tm * 16 + r) * DK + 2 * h;
      const float* bp = sKG + (tn * 16 + r) * DK + 2 * h;
      v8f accA = {}; v8f accQ = {};
      for (int kk = 0; kk < DK; kk += 4) {
        v2f bf = *(const v2f*)(bp + kk);
        accA = wmma4(*(const v2f*)(ap + kk), bf, accA);
        accQ = wmma4(*(const v2f*)(aq + kk), bf, accQ);
      }
      int gj = tn * 16 + r;
      for (int i2 = 0; i2 < 8; ++i2) {
        int gi = tm * 16 + i2 + 8 * h;
        float av = (gi > gj) ? sBeta[gi] * accA[i2] : ((gi == gj) ? 1.0f : 0.0f);
        float qv = (gi >= gj) ? accQ[i2] * scale : 0.0f;
        sA [gi * LCH + gj] = av;
        sQK[gi * LCH + gj] = qv;
      }
    }
    __syncthreads();

    // ---- T = A^-1 diag(beta): 64 independent forward substitutions ----
    if (tid < LCH) {
      int col = tid;
      for (int i = 0; i < LCH; ++i) {
        float s = (i == col) ? sBeta[i] : 0.0f;
        for (int j = col; j < i; ++j) s -= sA[i * LCH + j] * sT[j * LCH + col];
        sT[i * LCH + col] = s;
      }
    }
    __syncthreads();

    // ---- W = T @ kg (-> sQD, 64x128); U = T @ vg (-> sCR, 64x64) fused ----
    // tn == wave for the 32-tile loop, so the U predicate is wave-uniform and
    // EXEC stays all-ones inside every WMMA (ISA requirement).
    {
      int r = lane & 15, h = lane >> 4;
      bool doU = (wave < 4);
      for (int t = wave; t < 32; t += 8) {
        int tm = t >> 3, tn = t & 7;   // tn == wave
        const float* ap = sT  + (tm * 16 + r) * LCH + 2 * h;
        const float* bK = sKG + 2 * h * DK + tn * 16 + r;
        const float* bV = sV  + 2 * h * DVB + tn * 16 + r;   // only read when doU
        v8f accW = {}; v8f accU = {};
        if (doU) {
          for (int kk = 0; kk < LCH; kk += 4) {
            v2f a = *(const v2f*)(ap + kk);
            v2f bw; bw.x = bK[kk * DK];  bw.y = bK[kk * DK + DK];
            v2f bu; bu.x = bV[kk * DVB]; bu.y = bV[kk * DVB + DVB];
            accW = wmma4(a, bw, accW);
            accU = wmma4(a, bu, accU);
          }
        } else {
          for (int kk = 0; kk < LCH; kk += 4) {
            v2f a = *(const v2f*)(ap + kk);
            v2f bw; bw.x = bK[kk * DK]; bw.y = bK[kk * DK + DK];
            accW = wmma4(a, bw, accW);
          }
        }
        int gn = tn * 16 + r;
        for (int i2 = 0; i2 < 8; ++i2) {
          int gm = tm * 16 + i2 + 8 * h;
          sQD[gm * DK + gn] = accW[i2];
          if (doU) sCR[gm * DVB + gn] = accU[i2];
        }
      }
    }
    __syncthreads();

    // ---- corr = U - W @ S (16 tiles) ----
    for (int t = wave; t < 16; t += 8) {
      int tm = t >> 2, tn = t & 3;
      int r = lane & 15, h = lane >> 4;
      const float* ap = sQD + (tm * 16 + r) * DK + 2 * h;
      const float* bp = sS  + 2 * h * DVB + tn * 16 + r;
      v8f acc = {};
      for (int kk = 0; kk < DK; kk += 4) {
        v2f a = *(const v2f*)(ap + kk);
        v2f b; b.x = bp[kk * DVB]; b.y = bp[kk * DVB + DVB];
        acc = wmma4(a, b, acc);
      }
      int gn = tn * 16 + r;
      for (int i2 = 0; i2 < 8; ++i2) {
        int gm = tm * 16 + i2 + 8 * h;
        sCR[gm * DVB + gn] -= acc[i2];
      }
    }
    __syncthreads();

    // ---- O = (qn @ S)*exp(gmean)*scale + qk_masked @ corr -> global ----
    float* op = ob + (size_t)c * LCH * DV;
    for (int t = wave; t < 16; t += 8) {
      int tm = t >> 2, tn = t & 3;
      int r = lane & 15, h = lane >> 4;
      const float* ap = sQ + (tm * 16 + r) * DK + 2 * h;
      const float* bp = sS + 2 * h * DVB + tn * 16 + r;
      v8f acc = {};
      for (int kk = 0; kk < DK; kk += 4) {
        v2f a = *(const v2f*)(ap + kk);
        v2f b; b.x = bp[kk * DVB]; b.y = bp[kk * DVB + DVB];
        acc = wmma4(a, b, acc);
      }
      for (int i2 = 0; i2 < 8; ++i2)
        acc[i2] *= sGm[tm * 16 + i2 + 8 * h] * scale;
      const float* aq = sQK + (tm * 16 + r) * LCH + 2 * h;
      const float* bc = sCR + 2 * h * DVB + tn * 16 + r;
      for (int kk = 0; kk < LCH; kk += 4) {
        v2f a = *(const v2f*)(aq + kk);
        v2f b; b.x = bc[kk * DVB]; b.y = bc[kk * DVB + DVB];
        acc = wmma4(a, b, acc);
      }
      int gn = tn * 16 + r;
      for (int i2 = 0; i2 < 8; ++i2) {
        int gm = tm * 16 + i2 + 8 * h;
        op[(size_t)gm * DV + gn] = acc[i2];
      }
    }
    __syncthreads();

    // ---- S = diag(exp(g_total)) S + kg^T @ corr (32 tiles) ----
    for (int t = wave; t < 32; t += 8) {
      int tm = t >> 2, tn = t & 3;
      int r = lane & 15, h = lane >> 4;
      const float* at = sKG + 2 * h * DK + tm * 16 + r;   // transposed-A fragments
      const float* bc = sCR + 2 * h * DVB + tn * 16 + r;
      v8f acc = {};
      for (int kk = 0; kk < LCH; kk += 4) {
        v2f a; a.x = at[kk * DK];  a.y = at[kk * DK + DK];
        v2f b; b.x = bc[kk * DVB]; b.y = bc[kk * DVB + DVB];
        acc = wmma4(a, b, acc);
      }
      int gn = tn * 16 + r;
      for (int i2 = 0; i2 < 8; ++i2) {
        int gk = tm * 16 + i2 + 8 * h;
        sS[gk * DVB + gn] = sS[gk * DVB + gn] * sDecay[gk] + acc[i2];
      }
    }
    __syncthreads();  // protect staging buffers before next chunk's loads
  }

  // ---- emit final state ----
  float* st = state_out + (size_t)bh * DK * DV + vh * DVB;
  for (int i = tid; i < DK * DVB; i += NTHREADS) {
    int r = i >> 6, col = i & (DVB - 1);
    st[(size_t)r * DV + col] = sS[i];
  }
}

extern "C" void kernel_launch(void* const* d_in, const int* in_sizes, int n_in,
                              void* d_out, int out_size, void* d_ws, size_t ws_size,
                              hipStream_t stream) {
  const float* q    = (const float*)d_in[0];
  const float* k    = (const float*)d_in[1];
  const float* v    = (const float*)d_in[2];
  const float* g    = (const float*)d_in[3];
  const float* beta = (const float*)d_in[4];
  float* out   = (float*)d_out;                         // [B,H,S,V]
  float* state = out + (size_t)BHN * SEQ * DV;          // [B,H,K,V] appended
  kda_chunk_kernel<<<BHN * NVS, NTHREADS, 0, stream>>>(q, k, v, g, beta, out, state);
}